// SinkhornDistance_26671746908192
// MI455X (gfx1250) — compile-verified
//
#include <hip/hip_runtime.h>
#include <cstdint>
#include <cstddef>

// Problem constants (B=8, N=2048, eps=0.01, 50 fused u/v steps)
#define NN      2048
#define NB      8
#define EPSV    0.01f
#define K_E2    144.26950408889634f       // (1/eps) * log2(e)
#define EPS_LN2 0.006931471805599453f     // eps * ln(2)  (for eps*ln(s) = EPS_LN2*log2(s))
#define ELOGMU  (-0.07624618986159398f)   // eps * log(1/N) = -0.01*ln(2048)
#define SEG     8
#define NCHUNK  8
#define PITCH   2052                      // LDS row pitch (dwords): bank-conflict-free, 16B aligned
#define ITERS   50

typedef float v2f __attribute__((ext_vector_type(2)));
typedef float v8f __attribute__((ext_vector_type(8)));
typedef int   v4i __attribute__((ext_vector_type(4)));

typedef __attribute__((address_space(1))) v4i g_v4i;   // global (AS1) int4
typedef __attribute__((address_space(3))) v4i l_v4i;   // LDS    (AS3) int4

// Raw hardware exp2/log2: single v_exp_f32 / v_log_f32. Safe here: exp2 args are
// <= 0 (underflow contributes ~0 to sums >= 1); log2 args are >= 1.
#if __has_builtin(__builtin_amdgcn_exp2f)
__device__ __forceinline__ float fexp2(float x) { return __builtin_amdgcn_exp2f(x); }
#else
__device__ __forceinline__ float fexp2(float x) { return exp2f(x); }
#endif
#if __has_builtin(__builtin_amdgcn_logf)
__device__ __forceinline__ float flog2(float x) { return __builtin_amdgcn_logf(x); }
#else
__device__ __forceinline__ float flog2(float x) { return log2f(x); }
#endif

// Async global->LDS staging (CDNA5 GLOBAL_LOAD_ASYNC_TO_LDS_B128, ASYNCcnt)
#if __has_builtin(__builtin_amdgcn_global_load_async_to_lds_b128) && \
    __has_builtin(__builtin_amdgcn_s_wait_asynccnt)
#define USE_ASYNC_LDS 1
#endif

// ---------------------------------------------------------------------------
// Row pass: u[b,i] = eps*log_mu - eps*lse_j((c[b,i,j]+v[b,j])/eps)
// One block = 16 rows staged in LDS; 8 waves each own a 256-col chunk.
// Sum-of-exp accumulated on the matrix pipe via V_WMMA_F32_16X16X4_F32.
// ---------------------------------------------------------------------------
__global__ __launch_bounds__(256) void sk_row(const float* __restrict__ c,
                                              const float* __restrict__ v,
                                              float* __restrict__ u) {
    extern __shared__ float sm[];
    float* vsh  = sm;                    // NN floats
    float* slab = sm + NN;               // 16*PITCH floats
    float* mws  = slab + 16 * PITCH;     // 128: per-wave per-row max
    float* sws  = mws + 128;             // 128: per-wave per-row sum
    float* mrow = sws + 128;             // 16 : global row max

    const int tid  = threadIdx.x;
    const int wave = tid >> 5;
    const int lane = tid & 31;
    const int row  = lane & 15;          // WMMA A-layout: M = lane&15
    const int h    = lane >> 4;          // K-half: lanes 0-15 -> K0,K1; 16-31 -> K2,K3

    const int b    = blockIdx.x / (NN / 16);
    const int row0 = (blockIdx.x % (NN / 16)) * 16;

    // stage v[b,:] (broadcast operand)
    for (int k = tid; k < NN; k += 256) vsh[k] = v[b * NN + k];

    // stage 16xNN slab of c, fully coalesced 16B per lane
    const float* cbase = c + ((size_t)b * NN + row0) * NN;
#if USE_ASYNC_LDS
    #pragma unroll 4
    for (int it = 0; it < 32; ++it) {
        int f4  = tid + it * 256;            // float4 index in slab
        int r   = f4 >> 9;                   // / (NN/4)
        int col = (f4 & 511) << 2;
        __builtin_amdgcn_global_load_async_to_lds_b128(
            (g_v4i*)(cbase + (size_t)r * NN + col),
            (l_v4i*)(slab + r * PITCH + col),
            0, 0);
    }
    __builtin_amdgcn_s_wait_asynccnt(0);     // this wave's tiles resident
#else
    #pragma unroll 4
    for (int it = 0; it < 32; ++it) {
        int f4  = tid + it * 256;
        int r   = f4 >> 9;
        int col = (f4 & 511) << 2;
        float4 val = *reinterpret_cast<const float4*>(cbase + (size_t)r * NN + col);
        *reinterpret_cast<float4*>(slab + r * PITCH + col) = val;
    }
#endif
    __syncthreads();

    const int cb0 = wave * 256;

    // ---- phase 1: per-row max of x = c + v over this wave's chunk ----
    float m = -3.402823466e38f;
    #pragma unroll 4
    for (int s4 = 0; s4 < 64; ++s4) {
        int col = cb0 + s4 * 4 + 2 * h;
        v2f x  = *reinterpret_cast<const v2f*>(slab + row * PITCH + col);
        v2f vv = *reinterpret_cast<const v2f*>(vsh + col);
        m = fmaxf(m, fmaxf(x.x + vv.x, x.y + vv.y));
    }
    m = fmaxf(m, __shfl_xor(m, 16, 32));      // merge the two K-halves of each row
    if (h == 0) mws[wave * 16 + row] = m;
    __syncthreads();
    if (tid < 16) {
        float mm = mws[tid];
        #pragma unroll
        for (int w = 1; w < 8; ++w) mm = fmaxf(mm, mws[w * 16 + tid]);
        mrow[tid] = mm;
    }
    __syncthreads();

    const float mr = mrow[row];

    // ---- phase 2: sum exp2((x - m)*K) , reduced across K by WMMA ----
#if __has_builtin(__builtin_amdgcn_wmma_f32_16x16x4_f32)
    v8f acc = {0.f, 0.f, 0.f, 0.f, 0.f, 0.f, 0.f, 0.f};
    v2f bones = {1.0f, 1.0f};                 // B = 4x16 all-ones
    #pragma unroll 2
    for (int s4 = 0; s4 < 64; ++s4) {
        int col = cb0 + s4 * 4 + 2 * h;
        v2f x  = *reinterpret_cast<const v2f*>(slab + row * PITCH + col);
        v2f vv = *reinterpret_cast<const v2f*>(vsh + col);
        v2f a;
        a.x = fexp2((x.x + vv.x - mr) * K_E2);
        a.y = fexp2((x.y + vv.y - mr) * K_E2);
        // D[m][n] += sum_k A[m][k] : row sums replicated across N
        acc = __builtin_amdgcn_wmma_f32_16x16x4_f32(false, a, false, bones,
                                                    (short)0, acc, false, false);
    }
    // D layout: VGPR r: lanes 0-15 -> M=r ; lanes 16-31 -> M=r+8
    if (lane == 0) {
        #pragma unroll
        for (int r = 0; r < 8; ++r) sws[wave * 16 + r] = acc[r];
    } else if (lane == 16) {
        #pragma unroll
        for (int r = 0; r < 8; ++r) sws[wave * 16 + 8 + r] = acc[r];
    }
#else
    float sp = 0.f;
    #pragma unroll 2
    for (int s4 = 0; s4 < 64; ++s4) {
        int col = cb0 + s4 * 4 + 2 * h;
        v2f x  = *reinterpret_cast<const v2f*>(slab + row * PITCH + col);
        v2f vv = *reinterpret_cast<const v2f*>(vsh + col);
        sp += fexp2((x.x + vv.x - mr) * K_E2) + fexp2((x.y + vv.y - mr) * K_E2);
    }
    sp += __shfl_xor(sp, 16, 32);
    if (h == 0) sws[wave * 16 + row] = sp;
#endif
    __syncthreads();
    if (tid < 16) {
        float s = sws[tid];
        #pragma unroll
        for (int w = 1; w < 8; ++w) s += sws[w * 16 + tid];
        u[b * NN + row0 + tid] = ELOGMU - mrow[tid] - EPS_LN2 * flog2(s);
    }
}

// ---------------------------------------------------------------------------
// Col pass (partial): per-thread (max, sumexp) over a 256-row segment.
// Perfectly coalesced: block reads 256 consecutive floats per row.
// ---------------------------------------------------------------------------
__global__ __launch_bounds__(256) void sk_colpart(const float* __restrict__ c,
                                                  const float* __restrict__ u,
                                                  float2* __restrict__ part) {
    __shared__ float ush[256];
    const int tid = threadIdx.x;
    const int bx  = blockIdx.x;
    const int b   = bx >> 6;              // / (NCHUNK*SEG)
    const int r   = bx & 63;
    const int cch = r >> 3;
    const int seg = r & 7;
    const int j   = cch * 256 + tid;
    const int i0  = seg * 256;

    ush[tid] = u[b * NN + i0 + tid];
    __syncthreads();

    const float* cp = c + ((size_t)b * NN + i0) * NN + j;
    float m = -3.402823466e38f;
    #pragma unroll 8
    for (int i = 0; i < 256; ++i) m = fmaxf(m, cp[(size_t)i * NN] + ush[i]);
    float s = 0.f;
    #pragma unroll 4
    for (int i = 0; i < 256; ++i) s += fexp2((cp[(size_t)i * NN] + ush[i] - m) * K_E2);

    part[(size_t)(b * SEG + seg) * NN + j] = make_float2(m, s);
}

// Col pass (combine): merge 8 segment (m,s) pairs, finish v with clamp.
__global__ __launch_bounds__(256) void sk_colfin(const float2* __restrict__ part,
                                                 float* __restrict__ v) {
    const int t = blockIdx.x * 256 + threadIdx.x;   // B*N threads
    const int b = t >> 11;
    const int j = t & (NN - 1);
    float2 p[SEG];
    float m = -3.402823466e38f;
    #pragma unroll
    for (int k = 0; k < SEG; ++k) {
        p[k] = part[(size_t)(b * SEG + k) * NN + j];
        m = fmaxf(m, p[k].x);
    }
    float s = 0.f;
    #pragma unroll
    for (int k = 0; k < SEG; ++k) s += p[k].y * fexp2((p[k].x - m) * K_E2);
    float vv = ELOGMU - m - EPS_LN2 * flog2(s);
    if (vv > 9e8f) vv = 0.f;
    v[t] = vv;
}

// Final: pi = exp((c+u+v)/eps), negc = -c, vectorized float4.
__global__ __launch_bounds__(256) void sk_final(const float* __restrict__ c,
                                                const float* __restrict__ u,
                                                const float* __restrict__ v,
                                                float* __restrict__ pi,
                                                float* __restrict__ negc) {
    const size_t idx  = (size_t)blockIdx.x * 256 + threadIdx.x; // float4 index
    const size_t flat = idx * 4;
    const int b = (int)(flat >> 22);                 // / (N*N)
    const int rem = (int)(flat & ((1u << 22) - 1));
    const int i = rem >> 11;
    const int j = rem & (NN - 1);

    float4 c4 = *reinterpret_cast<const float4*>(c + flat);
    float  ui = u[b * NN + i];
    float4 v4 = *reinterpret_cast<const float4*>(v + b * NN + j);

    float4 p;
    p.x = fexp2((c4.x + ui + v4.x) * K_E2);
    p.y = fexp2((c4.y + ui + v4.y) * K_E2);
    p.z = fexp2((c4.z + ui + v4.z) * K_E2);
    p.w = fexp2((c4.w + ui + v4.w) * K_E2);
    *reinterpret_cast<float4*>(pi + flat) = p;

    float4 nc = make_float4(-c4.x, -c4.y, -c4.z, -c4.w);
    *reinterpret_cast<float4*>(negc + flat) = nc;
}

__global__ __launch_bounds__(256) void sk_zero(float* __restrict__ uv, int n) {
    int t = blockIdx.x * 256 + threadIdx.x;
    if (t < n) uv[t] = 0.f;
}

// ---------------------------------------------------------------------------
extern "C" void kernel_launch(void* const* d_in, const int* in_sizes, int n_in,
                              void* d_out, int out_size, void* d_ws, size_t ws_size,
                              hipStream_t stream) {
    (void)in_sizes; (void)n_in; (void)out_size; (void)ws_size;
    const float* c = (const float*)d_in[0];
    float* out = (float*)d_out;

    const size_t BNN = (size_t)NB * NN * NN;     // 33 554 432
    float* pi   = out;
    float* negc = out + BNN;
    float* u    = out + 2 * BNN;                 // working + final u
    float* v    = u + NB * NN;                   // working + final v
    float2* part = (float2*)d_ws;                // NB*SEG*NN float2 = 1 MiB

    constexpr int SMEM_FLOATS = NN + 16 * PITCH + 128 + 128 + 16;
    constexpr int SMEM_BYTES  = SMEM_FLOATS * (int)sizeof(float);  // ~137 KB (of 320 KB LDS)
    (void)hipFuncSetAttribute(reinterpret_cast<const void*>(sk_row),
                              hipFuncAttributeMaxDynamicSharedMemorySize, SMEM_BYTES);

    // u,v := 0 every call (deterministic; first row pass needs v = 0)
    const int nuv = 2 * NB * NN;
    sk_zero<<<(nuv + 255) / 256, 256, 0, stream>>>(u, nuv);

    for (int it = 0; it < ITERS; ++it) {
        sk_row<<<NB * (NN / 16), 256, SMEM_BYTES, stream>>>(c, v, u);
        sk_colpart<<<NB * NCHUNK * SEG, 256, 0, stream>>>(c, u, part);
        sk_colfin<<<(NB * NN) / 256, 256, 0, stream>>>(part, v);
    }

    sk_final<<<(unsigned)((BNN / 4) / 256), 256, 0, stream>>>(c, u, v, pi, negc);
}